// GCN_63462436765901
// MI455X (gfx1250) — compile-verified
//
#include <hip/hip_runtime.h>
#include <hip/hip_bf16.h>

#define NN    50000
#define EE    800000
#define HH    128
#define GG    512
#define FC1_  512
#define FC2_  256
#define CC    32

#define WT_STRIDE 130              // padded k-stride for LDS WT[n][k] (even, bank-spread)

typedef float v2f __attribute__((ext_vector_type(2)));
typedef float v8f __attribute__((ext_vector_type(8)));

#if defined(__HIP_DEVICE_COMPILE__) && __has_builtin(__builtin_amdgcn_global_load_async_to_lds_b32)
#define HAVE_ASYNC_LDS 1
typedef __attribute__((address_space(1))) int gas_int;   // global (device) addrspace
typedef __attribute__((address_space(3))) int las_int;   // LDS addrspace
#endif

// ---------------- degree / norm ----------------
__global__ void k_deg_init(float* deg) {
    int i = blockIdx.x * blockDim.x + threadIdx.x;
    if (i < NN) deg[i] = 1.0f;                 // self loop
}
__global__ void k_deg_count(const int* dst, float* deg) {
    int i = blockIdx.x * blockDim.x + threadIdx.x;
    if (i < EE) atomicAdd(&deg[dst[i]], 1.0f);
}
__global__ void k_deg_rsqrt(float* deg) {
    int i = blockIdx.x * blockDim.x + threadIdx.x;
    if (i < NN) deg[i] = rsqrtf(deg[i]);       // deg >= 1 always
}

// ---------------- node-layer GEMM: C[M,128] = A[M,128] @ B[128,128] ----------------
// Block: 256 threads = 8 waves. Block stages B transposed in LDS (async-to-LDS when
// available), each wave computes a 16x128 strip with 8 f32-WMMA accumulators,
// reusing one A fragment across 8 column tiles per K-step.
// __launch_bounds__(256, 1): allow a full VGPR budget (no spills) — one 8-wave
// workgroup per WGP is plenty for this WMMA-dense, LDS-fed kernel.
__global__ void __launch_bounds__(256, 1)
k_gcn_gemm(const float* __restrict__ A, const float* __restrict__ B,
           float* __restrict__ C, int M) {
    extern __shared__ float shW[];             // WT[n][k]: 128 * WT_STRIDE floats

    // stage: shW[n*WT_STRIDE + k] = B[k*128 + n]  (coalesced read, bank-spread write)
#ifdef HAVE_ASYNC_LDS
    for (int t = threadIdx.x; t < HH * HH; t += 256) {
        int k = t >> 7, n = t & 127;
        __builtin_amdgcn_global_load_async_to_lds_b32(
            (gas_int*)(unsigned long long)(B + t),
            (las_int*)(unsigned)(unsigned long long)&shW[n * WT_STRIDE + k],
            0, 0);
    }
    __builtin_amdgcn_s_wait_asynccnt(0);
    __syncthreads();
#else
    for (int t = threadIdx.x; t < HH * HH; t += 256) {
        int k = t >> 7, n = t & 127;
        shW[n * WT_STRIDE + k] = B[t];
    }
    __syncthreads();
#endif

    int wt = blockIdx.x * 8 + (threadIdx.x >> 5);      // 16-row tile index
    if (wt >= (M >> 4)) return;                        // wave-uniform
    int lane = threadIdx.x & 31;
    int half = lane >> 4;
    int mr   = lane & 15;

    v8f acc[8] = {};
    const float* Arow = A + (size_t)(wt * 16 + mr) * HH;

    for (int k0 = 0; k0 < HH; k0 += 4) {
        int ka = k0 + 2 * half;                        // K = vgpr + 2*half (ISA layout)
        v2f a;
        a[0] = Arow[ka];
        a[1] = Arow[ka + 1];
#pragma unroll
        for (int t = 0; t < 8; ++t) {
            v2f b = *(const v2f*)&shW[(t * 16 + mr) * WT_STRIDE + ka];  // ds_load_b64
            acc[t] = __builtin_amdgcn_wmma_f32_16x16x4_f32(
                         false, a, false, b, (short)0, acc[t], false, false);
        }
    }

#pragma unroll
    for (int t = 0; t < 8; ++t)
#pragma unroll
        for (int v = 0; v < 8; ++v)
            C[(size_t)(wt * 16 + v + 8 * half) * HH + t * 16 + mr] = acc[t][v];
}

// ---------------- generic fp32 WMMA GEMM (small FC layers) ----------------
// C[M,Nn] = A[M,K] @ B[K,Nn] (+bias) (+act: 0 none, 1 relu, 2 sigmoid)
__global__ void __launch_bounds__(256, 1)
k_wmma_gemm_f32(const float* __restrict__ A,
                const float* __restrict__ B,
                const float* __restrict__ bias,
                float* __restrict__ C,
                int M, int Nn, int K, int act) {
    int wave   = blockIdx.x * (blockDim.x >> 5) + (threadIdx.x >> 5);
    int tilesN = Nn >> 4;
    int tilesM = M >> 4;
    if (wave >= tilesM * tilesN) return;               // wave-uniform
    int mt = wave / tilesN;
    int nt = wave % tilesN;

    int lane = threadIdx.x & 31;
    int half = lane >> 4;
    int mr   = lane & 15;

    v8f acc = {0.f,0.f,0.f,0.f,0.f,0.f,0.f,0.f};
    const float* Arow = A + (size_t)(mt * 16 + mr) * K;
    const float* Bcol = B + nt * 16 + mr;

    for (int k0 = 0; k0 < K; k0 += 4) {
        int ka = k0 + 2 * half;
        v2f a, b;
        a[0] = Arow[ka];
        a[1] = Arow[ka + 1];
        b[0] = Bcol[(size_t)ka * Nn];
        b[1] = Bcol[(size_t)(ka + 1) * Nn];
        acc = __builtin_amdgcn_wmma_f32_16x16x4_f32(
                  false, a, false, b, (short)0, acc, false, false);
    }

    float bv = bias ? bias[nt * 16 + mr] : 0.0f;
#pragma unroll
    for (int v = 0; v < 8; ++v) {
        int m = mt * 16 + v + 8 * half;
        float val = acc[v] + bv;
        if (act == 1)      val = fmaxf(val, 0.0f);
        else if (act == 2) val = 1.0f / (1.0f + expf(-val));
        C[(size_t)m * Nn + nt * 16 + mr] = val;
    }
}

// ---------------- aggregation ----------------
__global__ void k_agg_self(const float* __restrict__ h, const float* __restrict__ dinv,
                           float* __restrict__ out) {
    int i = blockIdx.x * blockDim.x + threadIdx.x;      // over N*H/4
    if (i >= NN * (HH / 4)) return;
    int node = i / (HH / 4);
    float w = dinv[node]; w *= w;
    float4 v = ((const float4*)h)[i];
    float4 o; o.x = v.x * w; o.y = v.y * w; o.z = v.z * w; o.w = v.w * w;
    ((float4*)out)[i] = o;
}

// one wave per edge: out[dst] += h[src] * dinv[src]*dinv[dst]
__global__ void k_agg_edges(const float* __restrict__ h, const float* __restrict__ dinv,
                            const int* __restrict__ src, const int* __restrict__ dst,
                            float* __restrict__ out) {
    int wave = blockIdx.x * (blockDim.x >> 5) + (threadIdx.x >> 5);
    if (wave >= EE) return;
    int lane = threadIdx.x & 31;
    int s = src[wave];
    int d = dst[wave];
    float w = dinv[s] * dinv[d];
    float4 v = ((const float4*)(h + (size_t)s * HH))[lane];
    float* od = out + (size_t)d * HH + lane * 4;
    atomicAdd(od + 0, v.x * w);
    atomicAdd(od + 1, v.y * w);
    atomicAdd(od + 2, v.z * w);
    atomicAdd(od + 3, v.w * w);
}

__global__ void k_bias_relu(float* __restrict__ x, const float* __restrict__ b) {
    int i = blockIdx.x * blockDim.x + threadIdx.x;      // over N*H
    if (i >= NN * HH) return;
    x[i] = fmaxf(x[i] + b[i & (HH - 1)], 0.0f);
}

// ---------------- pooling ----------------
__global__ void k_zero(float* p, int n) {
    int i = blockIdx.x * blockDim.x + threadIdx.x;
    if (i < n) p[i] = 0.0f;
}
__global__ void k_pool(const float* __restrict__ x, const int* __restrict__ batch,
                       float* __restrict__ p) {
    int wave = blockIdx.x * (blockDim.x >> 5) + (threadIdx.x >> 5);   // one wave per node
    if (wave >= NN) return;
    int lane = threadIdx.x & 31;
    int g = batch[wave];
    float4 v = ((const float4*)(x + (size_t)wave * HH))[lane];
    float* pg = p + (size_t)g * HH + lane * 4;
    atomicAdd(pg + 0, v.x);
    atomicAdd(pg + 1, v.y);
    atomicAdd(pg + 2, v.z);
    atomicAdd(pg + 3, v.w);
}

// ---------------- batchnorm (training-mode batch stats, biased var) ----------------
__global__ void k_batchnorm(float* __restrict__ h, const float* __restrict__ gamma,
                            const float* __restrict__ beta, int cols) {
    int col = blockIdx.x * (blockDim.x >> 5) + (threadIdx.x >> 5);    // one wave per column
    if (col >= cols) return;
    int lane = threadIdx.x & 31;
    float sum = 0.0f, sumsq = 0.0f;
    for (int r = lane; r < GG; r += 32) {
        float v = h[(size_t)r * cols + col];
        sum += v; sumsq += v * v;
    }
    for (int off = 16; off > 0; off >>= 1) {
        sum   += __shfl_xor(sum, off, 32);
        sumsq += __shfl_xor(sumsq, off, 32);
    }
    float m   = sum * (1.0f / GG);
    float var = sumsq * (1.0f / GG) - m * m;
    float inv = rsqrtf(var + 1e-5f);
    float g = gamma[col], be = beta[col];
    for (int r = lane; r < GG; r += 32) {
        size_t idx = (size_t)r * cols + col;
        h[idx] = (h[idx] - m) * inv * g + be;
    }
}

// ---------------- host orchestration ----------------
extern "C" void kernel_launch(void* const* d_in, const int* in_sizes, int n_in,
                              void* d_out, int out_size, void* d_ws, size_t ws_size,
                              hipStream_t stream) {
    const float* x    = (const float*)d_in[0];
    const int*   ei   = (const int*)d_in[1];          // [2, E] row-major
    const int*   batch= (const int*)d_in[2];
    const float* W[4] = {(const float*)d_in[3], (const float*)d_in[5],
                         (const float*)d_in[7], (const float*)d_in[9]};
    const float* b[4] = {(const float*)d_in[4], (const float*)d_in[6],
                         (const float*)d_in[8], (const float*)d_in[10]};
    const float* lw1 = (const float*)d_in[11]; const float* lb1 = (const float*)d_in[12];
    const float* lw2 = (const float*)d_in[13]; const float* lb2 = (const float*)d_in[14];
    const float* lw3 = (const float*)d_in[15]; const float* lb3 = (const float*)d_in[16];
    const float* g1  = (const float*)d_in[17]; const float* be1 = (const float*)d_in[18];
    const float* g2  = (const float*)d_in[19]; const float* be2 = (const float*)d_in[20];

    const int* src = ei;            // edge_index[0]
    const int* dst = ei + EE;       // edge_index[1]

    // workspace carve-up (floats)
    float* dinv = (float*)d_ws;                         // N (padded to 50048)
    float* bufH = dinv + 50048;                         // N*H
    float* bufA = bufH + (size_t)NN * HH;               // N*H
    float* pool = bufA + (size_t)NN * HH;               // G*H
    float* fc1  = pool + (size_t)GG * HH;               // G*FC1
    float* fc2  = fc1  + (size_t)GG * FC1_;             // G*FC2
    float* out  = (float*)d_out;                        // G*C

    // --- structure: deg -> dinv ---
    k_deg_init <<<(NN + 255) / 256, 256, 0, stream>>>(dinv);
    k_deg_count<<<(EE + 255) / 256, 256, 0, stream>>>(dst, dinv);
    k_deg_rsqrt<<<(NN + 255) / 256, 256, 0, stream>>>(dinv);

    // --- 4 GCN layers ---
    const int rowTiles   = NN / 16;                     // 3125
    const int nodeBlocks = (rowTiles + 7) / 8;          // 8 waves (16-row strips) / block
    const size_t ldsW    = (size_t)HH * WT_STRIDE * sizeof(float);   // 66,560 B
    const float* cur = x;
    for (int l = 0; l < 4; ++l) {
        k_gcn_gemm <<<nodeBlocks, 256, ldsW, stream>>>(cur, W[l], bufH, NN);
        k_agg_self <<<(NN * (HH / 4) + 255) / 256, 256, 0, stream>>>(bufH, dinv, bufA);
        k_agg_edges<<<(EE + 7) / 8, 256, 0, stream>>>(bufH, dinv, src, dst, bufA);
        k_bias_relu<<<(NN * HH + 255) / 256, 256, 0, stream>>>(bufA, b[l]);
        cur = bufA;
    }

    // --- global add pool ---
    k_zero<<<(GG * HH + 255) / 256, 256, 0, stream>>>(pool, GG * HH);
    k_pool<<<(NN + 7) / 8, 256, 0, stream>>>(bufA, batch, pool);

    // --- MLP head ---
    {
        int tiles = (GG / 16) * (FC1_ / 16);
        k_wmma_gemm_f32<<<(tiles + 7) / 8, 256, 0, stream>>>(pool, lw1, lb1, fc1,
                                                             GG, FC1_, HH, 1);
    }
    k_batchnorm<<<(FC1_ + 7) / 8, 256, 0, stream>>>(fc1, g1, be1, FC1_);
    {
        int tiles = (GG / 16) * (FC2_ / 16);
        k_wmma_gemm_f32<<<(tiles + 7) / 8, 256, 0, stream>>>(fc1, lw2, lb2, fc2,
                                                             GG, FC2_, FC1_, 1);
    }
    k_batchnorm<<<(FC2_ + 7) / 8, 256, 0, stream>>>(fc2, g2, be2, FC2_);
    {
        int tiles = (GG / 16) * (CC / 16);
        k_wmma_gemm_f32<<<(tiles + 7) / 8, 256, 0, stream>>>(fc2, lw3, lb3, out,
                                                             GG, CC, FC2_, 2);
    }
}